// Model_1778116460930
// MI455X (gfx1250) — compile-verified
//
#include <hip/hip_runtime.h>
#include <hip/hip_bf16.h>

#define N_NODES 50000
#define N_EDGES 600000
#define FDIM    128          // F_IN == F == 128
#define OUT_DIM 12
#define NT      (N_NODES/16) // 3125 full 16-row tiles (50000 % 16 == 0)

#define RPB_XW  5            // M-tiles per block in k_xw  (3125 % 5 == 0)
#define RPB_ZR  2            // M-tiles per block in k_zr  (LDS: 3*2*8448B = 50.7KB)
#define RPB_H   3            // M-tiles per block in k_h   (LDS: 2*3*8448B = 50.7KB)

typedef float v2f __attribute__((ext_vector_type(2)));
typedef float v8f __attribute__((ext_vector_type(8)));

// Exact fp32 WMMA: D(16x16) = A(16x4) * B(4x16) + C
__device__ __forceinline__ v8f wmma4(v2f a, v2f b, v8f c) {
    return __builtin_amdgcn_wmma_f32_16x16x4_f32(
        false, a, false, b, (short)0, c, false, false);
}

// Async direct-to-LDS 16B copy (gfx1250 GLOBAL_LOAD_ASYNC_TO_LDS_B128).
// Generic __shared__ pointers carry the LDS byte offset in their low 32 bits
// (ISA aperture rule).
__device__ __forceinline__ void async_copy_b128(void* lds_dst, const void* gsrc) {
    unsigned lds_off = (unsigned)(unsigned long long)(uintptr_t)lds_dst;
    unsigned long long ga = (unsigned long long)(uintptr_t)gsrc;
    asm volatile("global_load_async_to_lds_b128 %0, %1, off"
                 :: "v"(lds_off), "v"(ga) : "memory");
}
__device__ __forceinline__ void async_wait0() {
    asm volatile("s_wait_asynccnt 0x0" ::: "memory");
}

// ---------------- degree / normalization ----------------

__global__ void k_deg_init(float* __restrict__ deg) {
    int i = blockIdx.x * blockDim.x + threadIdx.x;
    if (i < N_NODES) deg[i] = 1.0f;   // self-loop weight 1
}

__global__ void k_deg_acc(const long long* __restrict__ ei,
                          const float* __restrict__ ew,
                          float* __restrict__ deg) {
    int e = blockIdx.x * blockDim.x + threadIdx.x;
    if (e < N_EDGES) {
        long long d = ei[N_EDGES + e];     // dst row
        atomicAdd(&deg[d], ew[e]);
    }
}

__global__ void k_dinv(float* __restrict__ deg) {
    int i = blockIdx.x * blockDim.x + threadIdx.x;
    if (i < N_NODES) {
        float dv = deg[i];
        deg[i] = (dv > 0.f) ? rsqrtf(dv) : 0.f;
    }
}

// -------- fused XW3 = x @ [Wz|Wr|Wh]  (N x 384), 5 M-tiles per block --------

__global__ __launch_bounds__(256) void k_xw(
        const float* __restrict__ x,
        const float* __restrict__ Wz, const float* __restrict__ Wr,
        const float* __restrict__ Wh,
        float* __restrict__ XW3) {
    // Padded stride 132 floats: half-waves reading (k,k+1)/(k+2,k+3) hit
    // disjoint LDS bank sets; 528B row stride keeps 16B alignment.
    __shared__ float xs[RPB_XW][16][132];
    const int row0 = blockIdx.x * (16 * RPB_XW);
    for (int i = threadIdx.x; i < RPB_XW * 512; i += 256) {
        int t = i >> 9, s = i & 511;
        int r = s >> 5, c4 = s & 31;
        async_copy_b128(&xs[t][r][c4 * 4],
                        x + (size_t)(row0 + t * 16 + r) * FDIM + c4 * 4);
    }
    async_wait0();
    __syncthreads();

    const int lane = threadIdx.x & 31, wave = threadIdx.x >> 5;
    const int half = lane >> 4, l16 = lane & 15;
    const int col = wave * 16 + l16;                  // 0..127 within a gate

    v8f acc[3][RPB_XW];
    #pragma unroll
    for (int g = 0; g < 3; ++g)
        #pragma unroll
        for (int t = 0; t < RPB_XW; ++t) acc[g][t] = (v8f){};

    for (int k = 0; k < FDIM; k += 4) {
        int kk = k + 2 * half;
        v2f b0; b0.x = Wz[kk*FDIM + col]; b0.y = Wz[(kk+1)*FDIM + col];
        v2f b1; b1.x = Wr[kk*FDIM + col]; b1.y = Wr[(kk+1)*FDIM + col];
        v2f b2; b2.x = Wh[kk*FDIM + col]; b2.y = Wh[(kk+1)*FDIM + col];
        #pragma unroll
        for (int t = 0; t < RPB_XW; ++t) {
            v2f a; a.x = xs[t][l16][kk]; a.y = xs[t][l16][kk + 1];
            acc[0][t] = wmma4(a, b0, acc[0][t]);
            acc[1][t] = wmma4(a, b1, acc[1][t]);
            acc[2][t] = wmma4(a, b2, acc[2][t]);
        }
    }
    #pragma unroll
    for (int t = 0; t < RPB_XW; ++t) {
        #pragma unroll
        for (int v = 0; v < 8; ++v) {
            int row = row0 + t * 16 + v + 8 * half;
            float* o = XW3 + (size_t)row * 384;
            o[      col] = acc[0][t][v];
            o[128 + col] = acc[1][t][v];
            o[256 + col] = acc[2][t][v];
        }
    }
}

// ---------------- agg = bias + dinv^2 * XW3 (self-loop fused) ----------------

__global__ void k_agg_init(const float* __restrict__ XW3,
                           const float* __restrict__ dinv,
                           const float* __restrict__ bz,
                           const float* __restrict__ br,
                           const float* __restrict__ bh,
                           float* __restrict__ agg) {
    int idx = blockIdx.x * blockDim.x + threadIdx.x;  // float4 index
    const int TOT4 = N_NODES * 96;                    // 384/4 per row
    if (idx >= TOT4) return;
    int row = idx / 96;
    int colAbs = (idx % 96) * 4;                      // float4 never crosses gate
    int gate = colAbs >> 7;
    int f = colAbs & 127;
    const float* bp = (gate == 0) ? bz : (gate == 1) ? br : bh;
    float dv = dinv[row];
    float sn = dv * dv;
    float4 xv = ((const float4*)XW3)[idx];
    float4 o;
    o.x = bp[f+0] + sn * xv.x; o.y = bp[f+1] + sn * xv.y;
    o.z = bp[f+2] + sn * xv.z; o.w = bp[f+3] + sn * xv.w;
    ((float4*)agg)[idx] = o;
}

// ---------------- edge scatter: agg[dst] += norm_e * XW3[src] ----------------

__global__ __launch_bounds__(256) void k_scatter(
        const long long* __restrict__ ei, const float* __restrict__ ew,
        const float* __restrict__ dinv,
        const float* __restrict__ XW3, float* __restrict__ agg) {
    int wave = threadIdx.x >> 5, lane = threadIdx.x & 31;
    int e = blockIdx.x * 8 + wave;                    // one wave per edge
    if (e >= N_EDGES) return;
    long long s = ei[e];
    long long d = ei[N_EDGES + e];
    float nrm = dinv[s] * ew[e] * dinv[d];
    const float4* sp = (const float4*)(XW3 + (size_t)s * 384);
    float* dp = agg + (size_t)d * 384;
    #pragma unroll
    for (int j = 0; j < 3; ++j) {                     // 96 float4 over 32 lanes
        int f4 = lane + j * 32;
        float4 v = sp[f4];
        atomicAdd(dp + f4*4 + 0, nrm * v.x);
        atomicAdd(dp + f4*4 + 1, nrm * v.y);
        atomicAdd(dp + f4*4 + 2, nrm * v.z);
        atomicAdd(dp + f4*4 + 3, nrm * v.w);
    }
}

// -------- Z,R = sigmoid([agg_g | H] @ Lg + lbg), 2 M-tiles per block --------

__global__ __launch_bounds__(256) void k_zr(
        const float* __restrict__ agg, const float* __restrict__ H,
        const float* __restrict__ Lz, const float* __restrict__ lbz,
        const float* __restrict__ Lr, const float* __restrict__ lbr,
        float* __restrict__ Z, float* __restrict__ R) {
    __shared__ float azs[RPB_ZR][16][132], ars[RPB_ZR][16][132], hhs[RPB_ZR][16][132];
    const int tile0 = blockIdx.x * RPB_ZR;
    const int row0 = tile0 * 16;
    const int nth = (NT - tile0 < RPB_ZR) ? (NT - tile0) : RPB_ZR;
    for (int i = threadIdx.x; i < RPB_ZR * 512; i += 256) {
        int t = i >> 9, s = i & 511;
        int r = s >> 5, c4 = s & 31;
        if (t < nth) {
            const float* arow = agg + (size_t)(row0 + t * 16 + r) * 384;
            async_copy_b128(&azs[t][r][c4*4], arow + c4*4);
            async_copy_b128(&ars[t][r][c4*4], arow + 128 + c4*4);
            async_copy_b128(&hhs[t][r][c4*4],
                            H + (size_t)(row0 + t * 16 + r) * FDIM + c4*4);
        }
    }
    async_wait0();
    __syncthreads();

    const int lane = threadIdx.x & 31, wave = threadIdx.x >> 5;
    const int half = lane >> 4, l16 = lane & 15;
    const int col = wave * 16 + l16;

    v8f accz[RPB_ZR], accr[RPB_ZR];
    #pragma unroll
    for (int t = 0; t < RPB_ZR; ++t) { accz[t] = (v8f){}; accr[t] = (v8f){}; }

    for (int k = 0; k < FDIM; k += 4) {
        int kk = k + 2 * half;
        v2f bz_; bz_.x = Lz[kk*FDIM + col];        bz_.y = Lz[(kk+1)*FDIM + col];
        v2f br_; br_.x = Lr[kk*FDIM + col];        br_.y = Lr[(kk+1)*FDIM + col];
        v2f bz2; bz2.x = Lz[(FDIM+kk)*FDIM + col]; bz2.y = Lz[(FDIM+kk+1)*FDIM + col];
        v2f br2; br2.x = Lr[(FDIM+kk)*FDIM + col]; br2.y = Lr[(FDIM+kk+1)*FDIM + col];
        #pragma unroll
        for (int t = 0; t < RPB_ZR; ++t) {
            if (t < nth) {   // block-uniform branch: EXEC stays all-ones for WMMA
                v2f az; az.x = azs[t][l16][kk]; az.y = azs[t][l16][kk + 1];
                accz[t] = wmma4(az, bz_, accz[t]);
                v2f ar; ar.x = ars[t][l16][kk]; ar.y = ars[t][l16][kk + 1];
                accr[t] = wmma4(ar, br_, accr[t]);
                v2f ah; ah.x = hhs[t][l16][kk]; ah.y = hhs[t][l16][kk + 1];
                accz[t] = wmma4(ah, bz2, accz[t]);
                accr[t] = wmma4(ah, br2, accr[t]);
            }
        }
    }
    float biasz = lbz[col], biasr = lbr[col];
    #pragma unroll
    for (int t = 0; t < RPB_ZR; ++t) {
        if (t < nth) {
            #pragma unroll
            for (int v = 0; v < 8; ++v) {
                int row = row0 + t * 16 + v + 8 * half;
                size_t idx = (size_t)row * FDIM + col;
                Z[idx] = 1.f / (1.f + __expf(-(accz[t][v] + biasz)));
                R[idx] = 1.f / (1.f + __expf(-(accr[t][v] + biasr)));
            }
        }
    }
}

// ---- Ht = tanh([agg_h | H*R] @ Lh + lbh); h = Z*H+(1-Z)*Ht, 3 tiles/block ----

__global__ __launch_bounds__(256) void k_h(
        const float* __restrict__ agg, const float* __restrict__ H,
        const float* __restrict__ Z, const float* __restrict__ R,
        const float* __restrict__ Lh, const float* __restrict__ lbh,
        float* __restrict__ hout) {
    __shared__ float ahs[RPB_H][16][132], hrs[RPB_H][16][132];
    const int tile0 = blockIdx.x * RPB_H;
    const int row0 = tile0 * 16;
    const int nth = (NT - tile0 < RPB_H) ? (NT - tile0) : RPB_H;
    for (int i = threadIdx.x; i < RPB_H * 512; i += 256) {
        int t = i >> 9, s = i & 511;
        int r = s >> 5, c4 = s & 31;
        if (t < nth) {
            size_t rr = (size_t)(row0 + t * 16 + r);
            async_copy_b128(&ahs[t][r][c4*4], agg + rr * 384 + 256 + c4*4);
            float4 vh = ((const float4*)(H + rr * FDIM))[c4];
            float4 vr = ((const float4*)(R + rr * FDIM))[c4];
            hrs[t][r][c4*4+0] = vh.x * vr.x; hrs[t][r][c4*4+1] = vh.y * vr.y;
            hrs[t][r][c4*4+2] = vh.z * vr.z; hrs[t][r][c4*4+3] = vh.w * vr.w;
        }
    }
    async_wait0();
    __syncthreads();

    const int lane = threadIdx.x & 31, wave = threadIdx.x >> 5;
    const int half = lane >> 4, l16 = lane & 15;
    const int col = wave * 16 + l16;

    v8f acc[RPB_H];
    #pragma unroll
    for (int t = 0; t < RPB_H; ++t) acc[t] = (v8f){};

    for (int k = 0; k < FDIM; k += 4) {
        int kk = k + 2 * half;
        v2f b;  b.x  = Lh[kk*FDIM + col];        b.y  = Lh[(kk+1)*FDIM + col];
        v2f b2; b2.x = Lh[(FDIM+kk)*FDIM + col]; b2.y = Lh[(FDIM+kk+1)*FDIM + col];
        #pragma unroll
        for (int t = 0; t < RPB_H; ++t) {
            if (t < nth) {
                v2f a;  a.x  = ahs[t][l16][kk]; a.y  = ahs[t][l16][kk + 1];
                acc[t] = wmma4(a, b, acc[t]);
                v2f a2; a2.x = hrs[t][l16][kk]; a2.y = hrs[t][l16][kk + 1];
                acc[t] = wmma4(a2, b2, acc[t]);
            }
        }
    }
    float bias = lbh[col];
    #pragma unroll
    for (int t = 0; t < RPB_H; ++t) {
        if (t < nth) {
            #pragma unroll
            for (int v = 0; v < 8; ++v) {
                int row = row0 + t * 16 + v + 8 * half;
                size_t idx = (size_t)row * FDIM + col;
                float ht = tanhf(acc[t][v] + bias);
                float z  = Z[idx];
                float hp = H[idx];
                hout[idx] = z * hp + (1.f - z) * ht;
            }
        }
    }
}

// ---------------- pad Wo (128x12) to Wop (128x16) ----------------

__global__ void k_wop(const float* __restrict__ Wo, float* __restrict__ Wop) {
    int i = blockIdx.x * blockDim.x + threadIdx.x;
    if (i < FDIM * 16) {
        int k = i >> 4, n = i & 15;
        Wop[i] = (n < OUT_DIM) ? Wo[k * OUT_DIM + n] : 0.f;
    }
}

// ---------------- y = relu(h) @ Wo + bo ----------------

__global__ __launch_bounds__(256) void k_y(
        const float* __restrict__ hbuf, const float* __restrict__ Wop,
        const float* __restrict__ bo, float* __restrict__ y) {
    __shared__ float hs[8][16][132];
    const int wave = threadIdx.x >> 5, lane = threadIdx.x & 31;
    const int tile = blockIdx.x * 8 + wave;           // one 16-row tile / wave
    const bool valid = tile < NT;
    const int row0 = tile * 16;
    if (valid) {
        for (int r = 0; r < 16; ++r) {
            float4 v = ((const float4*)(hbuf + (size_t)(row0 + r) * FDIM))[lane];
            hs[wave][r][lane*4+0] = fmaxf(v.x, 0.f);
            hs[wave][r][lane*4+1] = fmaxf(v.y, 0.f);
            hs[wave][r][lane*4+2] = fmaxf(v.z, 0.f);
            hs[wave][r][lane*4+3] = fmaxf(v.w, 0.f);
        }
    }
    __syncthreads();
    if (!valid) return;

    const int half = lane >> 4, l16 = lane & 15;
    v8f acc = {};
    for (int k = 0; k < FDIM; k += 4) {
        int kk = k + 2 * half;
        v2f a; a.x = hs[wave][l16][kk]; a.y = hs[wave][l16][kk + 1];
        v2f b; b.x = Wop[kk*16 + l16]; b.y = Wop[(kk+1)*16 + l16];
        acc = wmma4(a, b, acc);
    }
    if (l16 < OUT_DIM) {
        float bias = bo[l16];
        #pragma unroll
        for (int v = 0; v < 8; ++v) {
            int row = row0 + v + 8 * half;
            y[(size_t)row * OUT_DIM + l16] = acc[v] + bias;
        }
    }
}

// ---------------- host launcher ----------------

extern "C" void kernel_launch(void* const* d_in, const int* in_sizes, int n_in,
                              void* d_out, int out_size, void* d_ws, size_t ws_size,
                              hipStream_t stream) {
    const float*     x   = (const float*)d_in[0];
    const long long* ei  = (const long long*)d_in[1];
    const float*     ew  = (const float*)d_in[2];
    const float*     H   = (const float*)d_in[3];
    const float*     Wz  = (const float*)d_in[4];
    const float*     bz  = (const float*)d_in[5];
    const float*     Wr  = (const float*)d_in[6];
    const float*     br  = (const float*)d_in[7];
    const float*     Wh  = (const float*)d_in[8];
    const float*     bh  = (const float*)d_in[9];
    const float*     Lz  = (const float*)d_in[10];
    const float*     lbz = (const float*)d_in[11];
    const float*     Lr  = (const float*)d_in[12];
    const float*     lbr = (const float*)d_in[13];
    const float*     Lh  = (const float*)d_in[14];
    const float*     lbh = (const float*)d_in[15];
    const float*     Wo  = (const float*)d_in[16];
    const float*     bo  = (const float*)d_in[17];

    float* y    = (float*)d_out;                          // [N, 12]
    float* hout = y + (size_t)N_NODES * OUT_DIM;          // [N, 128]

    float* ws   = (float*)d_ws;
    float* dinv = ws;                                     // N (deg -> dinv in place)
    float* XW3  = dinv + N_NODES;                         // N*384
    float* agg  = XW3 + (size_t)N_NODES * 384;            // N*384
    float* Zb   = agg + (size_t)N_NODES * 384;            // N*128
    float* Rb   = Zb  + (size_t)N_NODES * FDIM;           // N*128
    float* Wop  = Rb  + (size_t)N_NODES * FDIM;           // 128*16

    k_deg_init<<<(N_NODES + 255) / 256, 256, 0, stream>>>(dinv);
    k_deg_acc <<<(N_EDGES + 255) / 256, 256, 0, stream>>>(ei, ew, dinv);
    k_dinv    <<<(N_NODES + 255) / 256, 256, 0, stream>>>(dinv);

    k_xw      <<<NT / RPB_XW, 256, 0, stream>>>(x, Wz, Wr, Wh, XW3);
    k_agg_init<<<(N_NODES * 96 + 255) / 256, 256, 0, stream>>>(XW3, dinv, bz, br, bh, agg);
    k_scatter <<<(N_EDGES + 7) / 8, 256, 0, stream>>>(ei, ew, dinv, XW3, agg);

    k_zr      <<<(NT + RPB_ZR - 1) / RPB_ZR, 256, 0, stream>>>(agg, H, Lz, lbz, Lr, lbr, Zb, Rb);
    k_h       <<<(NT + RPB_H - 1) / RPB_H, 256, 0, stream>>>(agg, H, Zb, Rb, Lh, lbh, hout);

    k_wop     <<<(FDIM * 16 + 255) / 256, 256, 0, stream>>>(Wo, Wop);
    k_y       <<<(NT + 7) / 8, 256, 0, stream>>>(hout, Wop, bo, y);
}